// NoisyTopkRouter_40364102648331
// MI455X (gfx1250) — compile-verified
//
#include <hip/hip_runtime.h>
#include <math.h>

typedef __attribute__((ext_vector_type(2))) float v2f;
typedef __attribute__((ext_vector_type(8))) float v8f;

#define D_DIM 1024
#define E_DIM 8
#define N_DIM 16                 // 8 route cols + 8 noise cols fused
#define WAVES_PER_WG 8
#define TOKENS_PER_WAVE 16
#define TOKENS_PER_WG (WAVES_PER_WG * TOKENS_PER_WAVE)   // 128
#define WG_THREADS (WAVES_PER_WG * 32)                   // 256

__launch_bounds__(WG_THREADS)
__global__ void noisy_topk_router_kernel(const float* __restrict__ mh,     // [T, D]
                                         const float* __restrict__ Wr,     // [D, 8]
                                         const float* __restrict__ br,     // [8]
                                         const float* __restrict__ Wn,     // [D, 8]
                                         const float* __restrict__ bn,     // [8]
                                         const float* __restrict__ noise,  // [T, 8]
                                         const int*   __restrict__ train_p,
                                         float* __restrict__ out,          // [T,8] f32 then [T,2] i32
                                         int n_tokens) {
    __shared__ float lds_w[D_DIM * N_DIM];                                 // 64 KB (all weights)
    __shared__ float lds_tile[WAVES_PER_WG][TOKENS_PER_WAVE * N_DIM];      //  8 KB

    const int tid   = threadIdx.x;
    const int wave  = tid >> 5;
    const int lane  = tid & 31;
    const int half  = lane >> 4;      // K-split of the wave for A/B fragments
    const int m     = lane & 15;      // matrix row (A) / column (B,C,D)
    const int mlane = 2 * half;       // K offset owned by this half-wave

    const int tok      = blockIdx.x * TOKENS_PER_WG + wave * TOKENS_PER_WAVE + m;
    const int safe_tok = (tok < n_tokens) ? tok : (n_tokens - 1);  // keep EXEC full for WMMA
    const float* rowA  = mh + (size_t)safe_tok * D_DIM;

    // ---- Stage all fused weights once: lds_w[k*16 + n], n<8 route, n>=8 noise.
    // Task t: row k = t>>1, half h = t&1; 8 floats via two b128 ops.
    for (int t = tid; t < D_DIM * 2; t += WG_THREADS) {
        const int k = t >> 1;
        const int h = t & 1;
        const float4* src = (const float4*)((h ? Wn : Wr) + (size_t)k * E_DIM);
        const float4 lo = src[0];
        const float4 hi = src[1];
        float4* dst = (float4*)&lds_w[k * N_DIM + h * E_DIM];
        dst[0] = lo;
        dst[1] = hi;
    }
    __syncthreads();

    // ---- Software-pipelined WMMA loop: double-buffered A (global) and B (LDS)
    const float* wp = lds_w + mlane * N_DIM + m;   // B frag base: +kl*16 / +kl*16+16

    v2f a0 = *(const v2f*)(rowA + mlane + 0);
    v2f a1 = *(const v2f*)(rowA + mlane + 4);
    v2f a2 = *(const v2f*)(rowA + mlane + 8);
    v2f a3 = *(const v2f*)(rowA + mlane + 12);
    v2f b0 = { wp[0],   wp[16]  };
    v2f b1 = { wp[64],  wp[80]  };
    v2f b2 = { wp[128], wp[144] };
    v2f b3 = { wp[192], wp[208] };

    v8f acc = {};

    for (int kk = 0; kk < D_DIM; kk += 16) {
        const int kn = (kk + 16) & (D_DIM - 1);   // wrapped -> final preload stays in-bounds

        // issue next-batch loads (consumed next iteration)
        const float* ap = rowA + kn + mlane;
        const v2f na0 = *(const v2f*)(ap + 0);
        const v2f na1 = *(const v2f*)(ap + 4);
        const v2f na2 = *(const v2f*)(ap + 8);
        const v2f na3 = *(const v2f*)(ap + 12);
        const float* bp = wp + kn * N_DIM;
        const v2f nb0 = { bp[0],   bp[16]  };
        const v2f nb1 = { bp[64],  bp[80]  };
        const v2f nb2 = { bp[128], bp[144] };
        const v2f nb3 = { bp[192], bp[208] };

        // stream-ahead prefetch of the A row (speculative; OOB dropped)
        __builtin_prefetch(rowA + kk + 512, 0, 3);

        // consume current batch: 4 x V_WMMA_F32_16X16X4_F32
        acc = __builtin_amdgcn_wmma_f32_16x16x4_f32(false, a0, false, b0, (short)0, acc, false, false);
        acc = __builtin_amdgcn_wmma_f32_16x16x4_f32(false, a1, false, b1, (short)0, acc, false, false);
        acc = __builtin_amdgcn_wmma_f32_16x16x4_f32(false, a2, false, b2, (short)0, acc, false, false);
        acc = __builtin_amdgcn_wmma_f32_16x16x4_f32(false, a3, false, b3, (short)0, acc, false, false);

        a0 = na0; a1 = na1; a2 = na2; a3 = na3;
        b0 = nb0; b1 = nb1; b2 = nb2; b3 = nb3;
    }

    // Spill C tile: VGPR r holds row (8*half + r), column m
    #pragma unroll
    for (int r = 0; r < 8; ++r)
        lds_tile[wave][(half * 8 + r) * N_DIM + m] = acc[r];
    __syncthreads();

    // ---- Epilogue: lanes 0-15 of each wave each own one token
    if (lane < TOKENS_PER_WAVE && tok < n_tokens) {
        const int train = *train_p;
        float noisy[E_DIM];
        #pragma unroll
        for (int e = 0; e < E_DIM; ++e) {
            const float rl = lds_tile[wave][lane * N_DIM + e] + br[e];
            float v = rl;
            if (train) {
                const float nl = lds_tile[wave][lane * N_DIM + E_DIM + e] + bn[e];
                const float sp = (nl > 20.0f) ? nl : log1pf(__expf(nl)); // softplus
                v = rl + noise[(size_t)tok * E_DIM + e] * sp;
            }
            noisy[e] = v;
        }
        // top-2, lowest index wins on ties (matches jax.lax.top_k)
        int i0 = 0; float v0 = noisy[0];
        #pragma unroll
        for (int e = 1; e < E_DIM; ++e)
            if (noisy[e] > v0) { v0 = noisy[e]; i0 = e; }
        int i1 = -1; float v1 = -__builtin_inff();
        #pragma unroll
        for (int e = 0; e < E_DIM; ++e)
            if (e != i0 && noisy[e] > v1) { v1 = noisy[e]; i1 = e; }

        // softmax over {v0, v1}; masked (-inf) entries -> 0
        const float e1  = __expf(v1 - v0);
        const float inv = 1.0f / (1.0f + e1);
        const float p0 = inv, p1 = e1 * inv;

        float* orow = out + (size_t)tok * E_DIM;
        #pragma unroll
        for (int e = 0; e < E_DIM; ++e)
            orow[e] = (e == i0) ? p0 : ((e == i1) ? p1 : 0.0f);

        int* idx = (int*)(out + (size_t)n_tokens * E_DIM);
        idx[(size_t)tok * 2 + 0] = i0;
        idx[(size_t)tok * 2 + 1] = i1;
    }
}

extern "C" void kernel_launch(void* const* d_in, const int* in_sizes, int n_in,
                              void* d_out, int out_size, void* d_ws, size_t ws_size,
                              hipStream_t stream) {
    const float* mh    = (const float*)d_in[0];
    const float* Wr    = (const float*)d_in[1];
    const float* br    = (const float*)d_in[2];
    const float* Wn    = (const float*)d_in[3];
    const float* bn    = (const float*)d_in[4];
    const float* noise = (const float*)d_in[5];
    const int*   train = (const int*)d_in[6];
    float* out = (float*)d_out;

    const int n_tokens = in_sizes[5] / E_DIM;  // B*S from noise [B,S,E]
    const int grid = (n_tokens + TOKENS_PER_WG - 1) / TOKENS_PER_WG;

    noisy_topk_router_kernel<<<grid, WG_THREADS, 0, stream>>>(
        mh, Wr, br, Wn, bn, noise, train, out, n_tokens);
}